// HashSDFNetwork_19146964205939
// MI455X (gfx1250) — compile-verified
//
#include <hip/hip_runtime.h>
#include <hip/hip_bf16.h>

typedef __attribute__((ext_vector_type(2))) float v2f;
typedef __attribute__((ext_vector_type(8))) float v8f;

#define L_LVL   12
#define T_SIZE  (1u << 19)
#define TMASK   (T_SIZE - 1u)
#define NPTS    524288
#define DOUT    257
#define EMB     24
#define ROWS_PB 128        // points (rows) per block
#define ESTRIDE 26         // padded LDS stride (even -> 8B-aligned v2f rows)
#define WSTRIDE 26         // padded stride for transposed W tile [col][k]

// Fused Instant-NGP hash encode + [24 x 257] linear head.
// Phase 1: each of 128 threads encodes one point into LDS.
// Phase 2: 4 waves compute the 128x257 output tile with V_WMMA_F32_16X16X4_F32.
__global__ __launch_bounds__(128)
void ngp_fused_kernel(const float* __restrict__ in,      // [N,3]
                      const float* __restrict__ tables,  // [L,T,2]
                      const float* __restrict__ Wm,      // [24,257]
                      const float* __restrict__ bias,    // [257]
                      float* __restrict__ out)           // [N,257]
{
    __shared__ __align__(16) float sWt[DOUT * WSTRIDE];   // W transposed [col][k] ~26.1 KB
    __shared__ __align__(16) float sE[ROWS_PB * ESTRIDE]; // embeddings   ~13.3 KB
    __shared__ float sB[DOUT];                            // bias          ~1 KB

    const int tid = threadIdx.x;

    // ---- stage W transposed + bias into LDS ----
    for (int i = tid; i < EMB * DOUT; i += ROWS_PB) {
        const int k = i / DOUT;
        const int c = i - k * DOUT;
        sWt[c * WSTRIDE + k] = Wm[i];
    }
    for (int i = tid; i < DOUT; i += ROWS_PB) sB[i] = bias[i];

    // ---- phase 1: hash-grid encode one point per thread ----
    const float res[L_LVL] = {16.f, 24.f, 38.f, 60.f, 93.f, 145.f,
                              225.f, 350.f, 545.f, 847.f, 1317.f, 2048.f};
    const int p = blockIdx.x * ROWS_PB + tid;   // NPTS is an exact multiple of 128
    const float xx = in[3 * p + 0];
    const float yy = in[3 * p + 1];
    const float zz = in[3 * p + 2];

    #pragma unroll
    for (int l = 0; l < L_LVL; ++l) {
        const float px = xx * res[l], py = yy * res[l], pz = zz * res[l];
        const float flx = floorf(px), fly = floorf(py), flz = floorf(pz);
        const float gx = px - flx, gy = py - fly, gz = pz - flz;
        const unsigned ux = (unsigned)flx, uy = (unsigned)fly, uz = (unsigned)flz;
        const float* tl = tables + (size_t)l * T_SIZE * 2u;
        float a0 = 0.f, a1 = 0.f;
        #pragma unroll
        for (int i = 0; i < 8; ++i) {
            const unsigned cx = ux + ((i >> 2) & 1);
            const unsigned cy = uy + ((i >> 1) & 1);
            const unsigned cz = uz + (i & 1);
            const unsigned h = cx ^ (cy * 2654435761u) ^ (cz * 805459861u);
            const unsigned idx = h & TMASK;
            const float2 f = *reinterpret_cast<const float2*>(tl + (size_t)idx * 2u);
            const float w = ((i & 4) ? gx : 1.f - gx) *
                            ((i & 2) ? gy : 1.f - gy) *
                            ((i & 1) ? gz : 1.f - gz);
            a0 += w * f.x;
            a1 += w * f.y;
        }
        sE[tid * ESTRIDE + 2 * l + 0] = a0;
        sE[tid * ESTRIDE + 2 * l + 1] = a1;
    }

    __syncthreads();

    // ---- phase 2: [128x24] @ [24x257] + bias via V_WMMA_F32_16X16X4_F32 ----
    const int lane = tid & 31;
    const int wv   = tid >> 5;       // wave id 0..3 -> rows [wv*32, wv*32+32)
    const int hs   = lane >> 4;      // half-wave select (K-pair / row-half select)
    const int lm   = lane & 15;

    // Preload ALL A fragments (independent of column tile): 2 row tiles x 6 K-chunks.
    // ISA 16x4 f32 A layout: lane(0-15) K={0,1}, lane(16-31) K={2,3} -> offset hs*2.
    v2f afrag[2][6];
    #pragma unroll
    for (int rt = 0; rt < 2; ++rt) {
        const float* ap = &sE[(wv * 32 + rt * 16 + lm) * ESTRIDE + hs * 2];
        #pragma unroll
        for (int kc = 0; kc < 6; ++kc)
            afrag[rt][kc] = *reinterpret_cast<const v2f*>(ap + kc * 4);
    }

    const size_t rowg0 = (size_t)blockIdx.x * ROWS_PB + wv * 32 + hs * 8;

    // Keep this loop rolled: no cross-iteration reuse exists (A is hoisted), and a
    // compact body (~6 ds_load_b64 + 12 WMMA + 16 NT stores) keeps VGPR pressure
    // low so 8 blocks / 32 waves co-reside per WGP to hide gather+store latency.
    #pragma unroll 1
    for (int ct = 0; ct < 17; ++ct) {
        const int col  = ct * 16 + lm;
        const bool cok = (col < DOUT);        // data-select guard: EXEC stays all-1s
        const int colc = cok ? col : 0;       // clamped (garbage cols are never stored)
        const float bv = sB[colc];

        // B 4x16 fragment: VGPR j -> K = kc*4 + hs*2 + j. With W transposed in LDS,
        // the two K values are consecutive -> one aligned ds_load_b64 each.
        const float* bp = &sWt[colc * WSTRIDE + hs * 2];
        v2f bfrag[6];
        #pragma unroll
        for (int kc = 0; kc < 6; ++kc)
            bfrag[kc] = *reinterpret_cast<const v2f*>(bp + kc * 4);

        v8f c0 = {bv, bv, bv, bv, bv, bv, bv, bv};   // bias folded into accumulators
        v8f c1 = {bv, bv, bv, bv, bv, bv, bv, bv};
        #pragma unroll
        for (int kc = 0; kc < 6; ++kc) {             // two independent WMMA chains
            c0 = __builtin_amdgcn_wmma_f32_16x16x4_f32(
                     false, afrag[0][kc], false, bfrag[kc], (short)0, c0, false, false);
            c1 = __builtin_amdgcn_wmma_f32_16x16x4_f32(
                     false, afrag[1][kc], false, bfrag[kc], (short)0, c1, false, false);
        }

        if (cok) {
            // C/D layout: VGPR j -> M = j (lanes 0-15) or j+8 (lanes 16-31).
            // Non-temporal: 539 MB output stream must not evict L2-resident tables.
            #pragma unroll
            for (int j = 0; j < 8; ++j)
                __builtin_nontemporal_store(c0[j], &out[(rowg0 + j) * DOUT + col]);
            #pragma unroll
            for (int j = 0; j < 8; ++j)
                __builtin_nontemporal_store(c1[j], &out[(rowg0 + 16 + j) * DOUT + col]);
        }
    }
}

extern "C" void kernel_launch(void* const* d_in, const int* in_sizes, int n_in,
                              void* d_out, int out_size, void* d_ws, size_t ws_size,
                              hipStream_t stream) {
    const float* in     = (const float*)d_in[0];   // [N,3]
    const float* tables = (const float*)d_in[1];   // [L,T,F]
    const float* Wm     = (const float*)d_in[2];   // [24,257]
    const float* bias   = (const float*)d_in[3];   // [257]
    float* out = (float*)d_out;

    const int nblocks = NPTS / ROWS_PB;            // 4096
    hipLaunchKernelGGL(ngp_fused_kernel, dim3(nblocks), dim3(ROWS_PB), 0, stream,
                       in, tables, Wm, bias, out);
}